// RNNAssembly_52922587021429
// MI455X (gfx1250) — compile-verified
//
#include <hip/hip_runtime.h>
#include <math.h>

// ---------------------------------------------------------------------------
// Problem constants (match reference setup_inputs)
// ---------------------------------------------------------------------------
#define T_STEPS 512
#define BATCH   32
#define BHALF   16      // batch rows per scan workgroup (= WMMA M tile)
#define DIN     256
#define DOUT    256
#define NBLK    16
#define BLK     64
#define HDIM    1024
#define LDSROW  1028    // scan LDS row stride (floats): 4-bank skew per row
#define KC      64      // GEMM K-chunk (floats)
#define TSTR    68      // GEMM LDS tile row stride (floats): 4-bank skew

typedef float v2f __attribute__((ext_vector_type(2)));
typedef float v4f __attribute__((ext_vector_type(4)));
typedef float v8f __attribute__((ext_vector_type(8)));

// fp32 WMMA: D(16x16) = A(16x4) * B(4x16) + C  -> v_wmma_f32_16x16x4_f32
__device__ __forceinline__ v8f wmma_f32(v2f a, v2f b, v8f c) {
  return __builtin_amdgcn_wmma_f32_16x16x4_f32(
      /*neg_a=*/false, a, /*neg_b=*/false, b,
      /*c_mod=*/(short)0, c, /*reuse_a=*/false, /*reuse_b=*/false);
}

__device__ __forceinline__ float fast_tanh(float x) {
#if __has_builtin(__builtin_amdgcn_tanhf)
  return __builtin_amdgcn_tanhf(x);          // V_TANH_F32 (trans op)
#elif __has_builtin(__builtin_amdgcn_tanh_f32)
  return __builtin_amdgcn_tanh_f32(x);
#else
  return tanhf(x);
#endif
}

__device__ __forceinline__ v2f lo4(v4f v) { v2f r; r.x = v.x; r.y = v.y; return r; }
__device__ __forceinline__ v2f hi4(v4f v) { v2f r; r.x = v.z; r.y = v.w; return r; }

// Async global->LDS DMA of a rows x 64-float tile (b128 per lane, ASYNCcnt).
__device__ __forceinline__ void async_copy_tile(const float* __restrict__ gsrc,
                                                int gstride, unsigned ldsBaseFloats,
                                                int rows, int tid) {
  for (int idx = tid; idx < rows * 16; idx += 256) {
    const int r = idx >> 4, c4 = idx & 15;
    unsigned off = (ldsBaseFloats + (unsigned)(r * TSTR + c4 * 4)) * 4u;
    const float* gp = gsrc + (size_t)r * gstride + c4 * 4;
    asm volatile("global_load_async_to_lds_b128 %0, %1, off"
                 :: "v"(off), "v"(gp) : "memory");
  }
}

// ---------------------------------------------------------------------------
// NT GEMM:  D[m,n] = sum_k A[m,k] * W[n,k] (+ bias[n]).
// Workgroup (8 waves) computes 128x64; K chunked by 64 with double-buffered
// async-DMA LDS staging; inner loop is ds_read_b64 -> WMMA only.
// ---------------------------------------------------------------------------
__global__ __launch_bounds__(256)
void gemm_nt_wmma(const float* __restrict__ A, const float* __restrict__ W,
                  const float* __restrict__ bias, float* __restrict__ D,
                  int M, int N, int K) {
  extern __shared__ float lds[];               // A: 2x128xTSTR, W: 2x64xTSTR
  const unsigned W_OFF = 2u * 128 * TSTR;
  const int tid  = threadIdx.x;
  const int wave = tid >> 5;
  const int lane = tid & 31;
  const int row  = lane & 15;
  const int kh   = lane >> 4;
  const int m0   = blockIdx.y * 128;
  const int n0   = blockIdx.x * 64;

  v8f acc[4];
#pragma unroll
  for (int nt = 0; nt < 4; ++nt) acc[nt] = (v8f)0.0f;

  const int NC = K / KC;
  async_copy_tile(A + (size_t)m0 * K, K, 0u, 128, tid);
  async_copy_tile(W + (size_t)n0 * K, K, W_OFF, 64, tid);
  asm volatile("s_wait_asynccnt 0x0" ::: "memory");
  __syncthreads();

  // Per-lane LDS bases; all inner offsets are ds_read immediates.
  const float* __restrict__ aB = lds + (wave * 16 + row) * TSTR + 2 * kh;
  const float* __restrict__ w0 = lds + W_OFF + (row +  0) * TSTR + 2 * kh;
  const float* __restrict__ w1 = lds + W_OFF + (row + 16) * TSTR + 2 * kh;
  const float* __restrict__ w2 = lds + W_OFF + (row + 32) * TSTR + 2 * kh;
  const float* __restrict__ w3 = lds + W_OFF + (row + 48) * TSTR + 2 * kh;

  for (int c = 0; c < NC; ++c) {
    const int buf = c & 1;
    if (c + 1 < NC) {                          // prefetch next chunk via DMA
      const int nb = 1 - buf;
      async_copy_tile(A + (size_t)m0 * K + (c + 1) * KC, K,
                      (unsigned)(nb * 128 * TSTR), 128, tid);
      async_copy_tile(W + (size_t)n0 * K + (c + 1) * KC, K,
                      W_OFF + (unsigned)(nb * 64 * TSTR), 64, tid);
    }
    const int ab = buf * 128 * TSTR;
    const int wb = buf * 64 * TSTR;
#pragma unroll
    for (int kk = 0; kk < 16; ++kk) {
      const v2f a  = *(const v2f*)(aB + ab + kk * 4);
      const v2f b0 = *(const v2f*)(w0 + wb + kk * 4);
      const v2f b1 = *(const v2f*)(w1 + wb + kk * 4);
      const v2f b2 = *(const v2f*)(w2 + wb + kk * 4);
      const v2f b3 = *(const v2f*)(w3 + wb + kk * 4);
      acc[0] = wmma_f32(a, b0, acc[0]);
      acc[1] = wmma_f32(a, b1, acc[1]);
      acc[2] = wmma_f32(a, b2, acc[2]);
      acc[3] = wmma_f32(a, b3, acc[3]);
    }
    asm volatile("s_wait_asynccnt 0x0" ::: "memory");
    __syncthreads();
  }

#pragma unroll
  for (int nt = 0; nt < 4; ++nt) {
    const int n = n0 + nt * 16 + row;
    const float bv = bias ? bias[n] : 0.0f;
#pragma unroll
    for (int r = 0; r < 8; ++r) {
      const int m = m0 + wave * 16 + r + 8 * kh;   // C/D: VGPR r -> M = r | r+8
      D[(size_t)m * N + n] = acc[nt][r] + bv;
    }
  }
}

// ---------------------------------------------------------------------------
// One-shot weight repack: per (block n, k-step kk, lane) emit the 24 floats of
// WMMA B-fragments (Wb^T, K_n^T, -K_{n-1}) in load order, interleaved so each
// of the 6 v4f chunks is a fully-coalesced 512B wave read at an immediate
// offset. Sign of the -K_{i-1} coupling term is baked in here.
// ---------------------------------------------------------------------------
__global__ __launch_bounds__(256)
void pack_weights(const float* __restrict__ blocks, const float* __restrict__ coupl,
                  float* __restrict__ Wpk) {
  const int id   = blockIdx.x * 256 + threadIdx.x;  // 8192 threads total
  const int lane = id & 31;
  const int kk   = (id >> 5) & 15;
  const int n    = id >> 9;
  const int row  = lane & 15, kh = lane >> 4;
  const int nm1  = (n + NBLK - 1) & (NBLK - 1);
  const float* __restrict__ Wb = blocks + (size_t)n   * BLK * BLK;
  const float* __restrict__ Kn = coupl  + (size_t)n   * BLK * BLK;
  const float* __restrict__ Kp = coupl  + (size_t)nm1 * BLK * BLK;
  const int woff = row * BLK + 2 * kh;
  const int koff = 2 * kh * BLK + row;

  v2f bB[4], bK1[4], bK2[4];
#pragma unroll
  for (int nt = 0; nt < 4; ++nt) {
    bB [nt] = *(const v2f*)(Wb + woff + nt * 16 * BLK + kk * 4);
    bK1[nt] = *(const v2f*)(Kn + woff + nt * 16 * BLK + kk * 4);
    v2f t;
    t.x = -Kp[koff + kk * 4 * BLK + nt * 16];
    t.y = -Kp[koff + kk * 4 * BLK + nt * 16 + BLK];
    bK2[nt] = t;
  }
  float* dst = Wpk + (size_t)(n * 16 + kk) * (6 * 128) + lane * 4;
  v4f o;
  o.x = bB [0].x; o.y = bB [0].y; o.z = bB [1].x; o.w = bB [1].y; *(v4f*)(dst + 0 * 128) = o;
  o.x = bB [2].x; o.y = bB [2].y; o.z = bB [3].x; o.w = bB [3].y; *(v4f*)(dst + 1 * 128) = o;
  o.x = bK1[0].x; o.y = bK1[0].y; o.z = bK1[1].x; o.w = bK1[1].y; *(v4f*)(dst + 2 * 128) = o;
  o.x = bK1[2].x; o.y = bK1[2].y; o.z = bK1[3].x; o.w = bK1[3].y; *(v4f*)(dst + 3 * 128) = o;
  o.x = bK2[0].x; o.y = bK2[0].y; o.z = bK2[1].x; o.w = bK2[1].y; *(v4f*)(dst + 4 * 128) = o;
  o.x = bK2[2].x; o.y = bK2[2].y; o.z = bK2[3].x; o.w = bK2[3].y; *(v4f*)(dst + 5 * 128) = o;
}

// ---------------------------------------------------------------------------
// Sequential scan. grid = 2 workgroups (one per 16-batch half), 512 threads
// = 16 waves; wave n owns ring block n. State [16 x 1024] fp32 ping-pongs in
// LDS. Per step per wave: 192 v_wmma_f32_16x16x4_f32; packed weight frags
// streamed as 6 coalesced b128 loads per k-step with a depth-2 pipeline;
// P/G loads for the step issued up front to hide under the WMMA chain.
// ---------------------------------------------------------------------------
struct Frags { v4f c[6]; };

__device__ __forceinline__ void load_frags(Frags& f, const float* __restrict__ wp,
                                           int kk) {
#pragma unroll
  for (int j = 0; j < 6; ++j)
    f.c[j] = *(const v4f*)(wp + kk * (6 * 128) + j * 128);
}

__global__ __launch_bounds__(512)
void scan_kernel(const float* __restrict__ P,          // [T,32,H] input proj
                 const float* __restrict__ G,          // [T,32,H] gate proj+b
                 const float* __restrict__ init_state, // [32,H]
                 const float* __restrict__ Wpk,        // packed weight frags
                 const float* __restrict__ gate_state, // [H]
                 float* __restrict__ states) {         // [T,32,H]
  extern __shared__ float lds[];                       // 2 * BHALF * LDSROW
  const int wg   = blockIdx.x;        // batch half: rows [wg*16, wg*16+16)
  const int tid  = threadIdx.x;
  const int n    = tid >> 5;          // wave index == ring block index
  const int lane = tid & 31;
  const int row  = lane & 15;
  const int kh   = lane >> 4;
  const int np1  = (n + 1) & (NBLK - 1);
  const int nm1  = (n + NBLK - 1) & (NBLK - 1);

  const float* __restrict__ wp = Wpk + (size_t)n * (16 * 6 * 128) + lane * 4;
  const int aoff  = row * LDSROW + 2 * kh;                // LDS A-frag base
  const int lbase = 8 * kh * LDSROW + n * BLK + row;      // LDS C/D base

  float gs[4];
#pragma unroll
  for (int nt = 0; nt < 4; ++nt) gs[nt] = gate_state[n * BLK + nt * 16 + row];

  // Initial state -> LDS buffer 0 via async global->LDS DMA (ASYNCcnt).
  for (int idx = tid; idx < BHALF * (HDIM / 4); idx += 512) {
    const int r = idx >> 8;             // 256 float4 per batch row
    const int c = (idx & 255) << 2;     // float column
    unsigned lds_off = (unsigned)((r * LDSROW + c) * 4);
    const float* gp = init_state + (size_t)(wg * BHALF + r) * HDIM + c;
    asm volatile("global_load_async_to_lds_b128 %0, %1, off"
                 :: "v"(lds_off), "v"(gp) : "memory");
  }
  asm volatile("s_wait_asynccnt 0x0" ::: "memory");
  __syncthreads();

  const int BUF = BHALF * LDSROW;
  for (int t = 0; t < T_STEPS; ++t) {
    const float* __restrict__ Sc = lds + (t & 1) * BUF;
    float* __restrict__ Sn = lds + ((t + 1) & 1) * BUF;

    // Issue this step's P/G loads first: 64 loads in flight under the WMMAs.
    const size_t tbase =
        ((size_t)t * BATCH + wg * BHALF + 8 * kh) * HDIM + n * BLK + row;
    const float* __restrict__ Pb = P + tbase;
    const float* __restrict__ Gb = G + tbase;
    float pv[4][8], gv[4][8];
#pragma unroll
    for (int nt = 0; nt < 4; ++nt)
#pragma unroll
      for (int r = 0; r < 8; ++r) {
        pv[nt][r] = Pb[nt * 16 + r * HDIM];
        gv[nt][r] = Gb[nt * 16 + r * HDIM];
      }

    v8f acc[4];
#pragma unroll
    for (int nt = 0; nt < 4; ++nt) acc[nt] = (v8f)0.0f;

    // Depth-2 pipelined packed-weight streaming over the 16 k-steps.
    Frags cur, nxt;
    load_frags(cur, wp, 0);
#pragma unroll
    for (int kk = 0; kk < 16; ++kk) {
      if (kk < 15) load_frags(nxt, wp, kk + 1);
      v2f a0 = *(const v2f*)(Sc + aoff + n * BLK + kk * 4);     // tanh path
      a0.x = fast_tanh(a0.x); a0.y = fast_tanh(a0.y);
      const v2f a1 = *(const v2f*)(Sc + aoff + np1 * BLK + kk * 4);
      const v2f a2 = *(const v2f*)(Sc + aoff + nm1 * BLK + kk * 4); // sign baked
      acc[0] = wmma_f32(a0, lo4(cur.c[0]), acc[0]);
      acc[1] = wmma_f32(a0, hi4(cur.c[0]), acc[1]);
      acc[2] = wmma_f32(a0, lo4(cur.c[1]), acc[2]);
      acc[3] = wmma_f32(a0, hi4(cur.c[1]), acc[3]);
      acc[0] = wmma_f32(a1, lo4(cur.c[2]), acc[0]);
      acc[1] = wmma_f32(a1, hi4(cur.c[2]), acc[1]);
      acc[2] = wmma_f32(a1, lo4(cur.c[3]), acc[2]);
      acc[3] = wmma_f32(a1, hi4(cur.c[3]), acc[3]);
      acc[0] = wmma_f32(a2, lo4(cur.c[4]), acc[0]);
      acc[1] = wmma_f32(a2, hi4(cur.c[4]), acc[1]);
      acc[2] = wmma_f32(a2, lo4(cur.c[5]), acc[2]);
      acc[3] = wmma_f32(a2, hi4(cur.c[5]), acc[3]);
      if (kk < 15) cur = nxt;
    }

    // Prefetch next timestep's projection rows (global_prefetch_b8).
    if (t + 1 < T_STEPS) {
      __builtin_prefetch(Pb + (size_t)BATCH * HDIM, 0, 0);
      __builtin_prefetch(Gb + (size_t)BATCH * HDIM, 0, 0);
    }

    // Gated Euler update + writeback (global states + LDS ping-pong).
    float* __restrict__ Sb = states + tbase;
#pragma unroll
    for (int nt = 0; nt < 4; ++nt) {
#pragma unroll
      for (int r = 0; r < 8; ++r) {
        const float s = Sc[lbase + r * LDSROW + nt * 16];
        const float z = gv[nt][r] + gs[nt] * s;
        const float gate = 1.0f / (1.0f + __expf(-z));
        const float step = 1e-4f + gate * (0.1f - 1e-4f);
        const float ns = s + step * (acc[nt][r] - s + pv[nt][r]);
        Sn[lbase + r * LDSROW + nt * 16] = ns;
        Sb[nt * 16 + r * HDIM] = ns;
      }
    }
    __syncthreads();
  }
}

// ---------------------------------------------------------------------------
// Launch: pack -> proj GEMMs (P, G) -> scan -> output GEMM, all on `stream`.
// d_ws: P (64 MB) | G (64 MB) | packed weights (768 KB).
// d_out: output [T,B,256] then states [T,B,H].
// ---------------------------------------------------------------------------
extern "C" void kernel_launch(void* const* d_in, const int* in_sizes, int n_in,
                              void* d_out, int out_size, void* d_ws, size_t ws_size,
                              hipStream_t stream) {
  const float* input      = (const float*)d_in[0]; // [T,B,Din]
  const float* init_state = (const float*)d_in[1]; // [B,H]
  const float* blocks     = (const float*)d_in[2]; // [16,64,64]
  const float* coupling   = (const float*)d_in[3]; // [16,64,64]
  const float* W_in       = (const float*)d_in[4]; // [H,Din]
  const float* W_out      = (const float*)d_in[5]; // [Dout,H]
  const float* gate_W     = (const float*)d_in[6]; // [H,Din]
  const float* gate_b     = (const float*)d_in[7]; // [H]
  const float* gate_state = (const float*)d_in[8]; // [H]

  float* out_proj = (float*)d_out;                             // [T,B,Dout]
  float* states   = out_proj + (size_t)T_STEPS * BATCH * DOUT; // [T,B,H]

  float* P   = (float*)d_ws;                                   // [T*B, H]
  float* G   = P + (size_t)T_STEPS * BATCH * HDIM;             // [T*B, H]
  float* Wpk = G + (size_t)T_STEPS * BATCH * HDIM;             // 196608 floats

  const int M = T_STEPS * BATCH;                               // 16384

  // One-shot WMMA-fragment weight repack (16*16*32 lanes).
  pack_weights<<<dim3(32), 256, 0, stream>>>(blocks, coupling, Wpk);

  const size_t gemm_lds = (2u * 128 * TSTR + 2u * 64 * TSTR) * sizeof(float);
  (void)hipFuncSetAttribute((const void*)gemm_nt_wmma,
                            hipFuncAttributeMaxDynamicSharedMemorySize,
                            (int)gemm_lds);

  // P = input @ W_in^T   (M x H, K = Din)
  gemm_nt_wmma<<<dim3(HDIM / 64, M / 128), 256, gemm_lds, stream>>>(
      input, W_in, nullptr, P, M, HDIM, DIN);
  // G = input @ gate_W^T + gate_b
  gemm_nt_wmma<<<dim3(HDIM / 64, M / 128), 256, gemm_lds, stream>>>(
      input, gate_W, gate_b, G, M, HDIM, DIN);

  // Sequential scan: 2 workgroups, 16 waves each, ~128.5 KB dynamic LDS.
  const size_t scan_lds = 2u * BHALF * LDSROW * sizeof(float);
  (void)hipFuncSetAttribute((const void*)scan_kernel,
                            hipFuncAttributeMaxDynamicSharedMemorySize,
                            (int)scan_lds);
  scan_kernel<<<dim3(2), dim3(512), scan_lds, stream>>>(
      P, G, init_state, Wpk, gate_state, states);

  // output = states @ W_out^T   (M x Dout, K = H)
  gemm_nt_wmma<<<dim3(DOUT / 64, M / 128), 256, gemm_lds, stream>>>(
      states, W_out, nullptr, out_proj, M, DOUT, HDIM);
}